// CompGraphConv_47622597378119
// MI455X (gfx1250) — compile-verified
//
#include <hip/hip_runtime.h>

typedef __attribute__((ext_vector_type(2))) float v2f;
typedef __attribute__((ext_vector_type(4))) float v4f;
typedef __attribute__((ext_vector_type(8))) float v8f;

#define D 64

// ---------------- zero workspace ----------------
__global__ void zero_ws_kernel(float* __restrict__ p, size_t ntot) {
    size_t n4 = ntot >> 2;
    size_t i = (size_t)blockIdx.x * blockDim.x + threadIdx.x;
    size_t stride = (size_t)gridDim.x * blockDim.x;
    v4f z = {0.f, 0.f, 0.f, 0.f};
    for (size_t j = i; j < n4; j += stride) ((v4f*)p)[j] = z;
    if (i == 0) for (size_t j = (n4 << 2); j < ntot; ++j) p[j] = 0.f;
}

// ---------------- edge scatter: sum[half][dst] += x[src], deg[half][dst] += 1 ----------------
__global__ __launch_bounds__(256)
void edge_scatter_kernel(const float* __restrict__ x,
                         const int* __restrict__ src, const int* __restrict__ dst,
                         float* __restrict__ sumO, float* __restrict__ sumI,
                         float* __restrict__ degO, float* __restrict__ degI,
                         int E, int half) {
    unsigned gid = blockIdx.x * blockDim.x + threadIdx.x;
    int e = (int)(gid >> 5);           // one wave32 per edge
    int lane = threadIdx.x & 31;
    if (e >= E) return;
    int s = __builtin_amdgcn_readfirstlane(src[e]);
    int d = __builtin_amdgcn_readfirstlane(dst[e]);
    v2f v = *(const v2f*)(x + (size_t)s * D + lane * 2);   // 2 feats per lane
    bool firstHalf = (e < half);
    float* sum = firstHalf ? sumO : sumI;
    float* deg = firstHalf ? degO : degI;
    float* b = sum + (size_t)d * D + lane * 2;
    unsafeAtomicAdd(b,     v.x);       // global_atomic_add_f32 (L2-resident)
    unsafeAtomicAdd(b + 1, v.y);
    if (lane == 0) unsafeAtomicAdd(deg + d, 1.0f);
}

// ---------------- node GEMM: 16-node tile, 3 fused K=64 matmuls via f32 WMMA ----------------
__global__ __launch_bounds__(128)
void node_gemm_kernel(const float* __restrict__ x, const float* __restrict__ r,
                      const float* __restrict__ sumO, const float* __restrict__ sumI,
                      const float* __restrict__ degO, const float* __restrict__ degI,
                      const float* __restrict__ W_S, const float* __restrict__ b_S,
                      const float* __restrict__ W_O, const float* __restrict__ b_O,
                      const float* __restrict__ W_I, const float* __restrict__ b_I,
                      float* __restrict__ out, int N) {
    __shared__ float A[16][200];      // fused A: [x-r | sumO-degO*x | sumI-degI*x], padded
    __shared__ float sdO[16], sdI[16];

    const int tid  = threadIdx.x;
    const int base = blockIdx.x * 16;

    // Stage fused A tile (16 rows x 192 cols) into LDS
    for (int idx = tid; idx < 16 * D; idx += 128) {
        int m = idx >> 6, k = idx & (D - 1);
        int v = base + m;
        float xv = 0.f, sO = 0.f, sI = 0.f, dO = 0.f, dI = 0.f, a0 = 0.f;
        if (v < N) {
            size_t o = (size_t)v * D + k;
            xv = x[o]; sO = sumO[o]; sI = sumI[o];
            dO = degO[v]; dI = degI[v];
            a0 = xv - r[k];
        }
        A[m][k]           = a0;
        A[m][D + k]       = sO - dO * xv;
        A[m][2 * D + k]   = sI - dI * xv;
    }
    if (tid < 16) {
        int v = base + tid;
        sdO[tid] = (v < N) ? degO[v] : 0.f;
        sdI[tid] = (v < N) ? degI[v] : 0.f;
    }
    __syncthreads();

    const int lane  = tid & 31;
    const int wave  = tid >> 5;          // 4 waves -> 4 x 16 output columns
    const int n0    = wave * 16;
    const int ncol  = n0 + (lane & 15);  // this lane's output column
    const int khalf = (lane >> 4) * 2;   // lanes 16-31 hold K+2, K+3
    const int arow  = lane & 15;

    v8f c = {};
    const float* Wmat[3] = {W_S, W_O, W_I};
    const int    aoff[3] = {0, D, 2 * D};

#pragma unroll
    for (int mat = 0; mat < 3; ++mat) {
        const float* W = Wmat[mat];
        // B fragments: B[k][n] = W[n][k]; lane holds (W[n,kb], W[n,kb+1]), kb = kk*4+khalf
        v2f bf[16];
#pragma unroll
        for (int kk = 0; kk < 16; ++kk)
            bf[kk] = *(const v2f*)(W + (size_t)ncol * D + kk * 4 + khalf);
#pragma unroll
        for (int kk = 0; kk < 16; ++kk) {
            v2f a = *(const v2f*)&A[arow][aoff[mat] + kk * 4 + khalf];
            c = __builtin_amdgcn_wmma_f32_16x16x4_f32(
                    false, a, false, bf[kk], (short)0, c, false, false);
        }
    }

    // Epilogue: C/D layout -> VGPR rr holds M = rr + (lane>=16 ? 8 : 0)
    float bs = b_S[ncol], bo = b_O[ncol], bi = b_I[ncol];
    const int mhi = (lane >> 4) * 8;
#pragma unroll
    for (int rr = 0; rr < 8; ++rr) {
        int m = rr + mhi;
        int v = base + m;
        if (v < N)
            out[(size_t)v * D + ncol] = c[rr] + bs + sdO[m] * bo + sdI[m] * bi;
    }
}

// ---------------- r_out = r @ W_R^T + b_R (64 elements) ----------------
__global__ void rout_kernel(const float* __restrict__ r, const float* __restrict__ W_R,
                            const float* __restrict__ b_R, float* __restrict__ out) {
    int n = threadIdx.x;
    if (n < D) {
        float acc = b_R[n];
        for (int k = 0; k < D; ++k) acc += r[k] * W_R[n * D + k];
        out[n] = acc;
    }
}

extern "C" void kernel_launch(void* const* d_in, const int* in_sizes, int n_in,
                              void* d_out, int out_size, void* d_ws, size_t ws_size,
                              hipStream_t stream) {
    const float* x    = (const float*)d_in[0];
    const float* r    = (const float*)d_in[1];
    const int*   src  = (const int*)d_in[2];
    const int*   dst  = (const int*)d_in[3];
    const float* W_O  = (const float*)d_in[4];
    const float* b_O  = (const float*)d_in[5];
    const float* W_I  = (const float*)d_in[6];
    const float* b_I  = (const float*)d_in[7];
    const float* W_S  = (const float*)d_in[8];
    const float* b_S  = (const float*)d_in[9];
    const float* W_R  = (const float*)d_in[10];
    const float* b_R  = (const float*)d_in[11];

    const int N = in_sizes[0] / D;
    const int E = in_sizes[2];
    const int half = E / 2;

    float* sumO = (float*)d_ws;
    float* sumI = sumO + (size_t)N * D;
    float* degO = sumI + (size_t)N * D;
    float* degI = degO + N;
    float* out  = (float*)d_out;

    size_t ntot = (size_t)N * (2 * D + 2);
    zero_ws_kernel<<<2048, 256, 0, stream>>>(sumO, ntot);

    int edgeBlocks = (E + 7) / 8;   // 8 waves (edges) per 256-thread block
    edge_scatter_kernel<<<edgeBlocks, 256, 0, stream>>>(x, src, dst, sumO, sumI,
                                                        degO, degI, E, half);

    int nodeBlocks = (N + 15) / 16;
    node_gemm_kernel<<<nodeBlocks, 128, 0, stream>>>(x, r, sumO, sumI, degO, degI,
                                                     W_S, b_S, W_O, b_O, W_I, b_I,
                                                     out, N);

    rout_kernel<<<1, 64, 0, stream>>>(r, W_R, b_R, out + (size_t)N * D);
}